// BernConv_6124623364542
// MI455X (gfx1250) — compile-verified
//
#include <hip/hip_runtime.h>

// BernConv K=2:  out = c0*f2 + c1*(f2 - Ahat f2),  f2 = (I+Ahat)^2 feat
//   c0 = 0.25*relu(w0),  c1 = 0.5*relu(w1) + 0.25*relu(w2)
// Ahat x = dinv .* segsum( (x .* dinv)[src], dst ),  dinv = max(indeg,1)^-0.5
//
// Whole working set (~26 MB) is resident in the 192 MB L2 -> L2-bound scatter.
// Wave32-native layout: one wave per edge, 32 lanes = 32 features = one 128B line.

#define D_FEAT 32

__global__ __launch_bounds__(256) void zero4_kernel(float4* __restrict__ p, int n4) {
    int i = blockIdx.x * blockDim.x + threadIdx.x;
    if (i < n4) p[i] = make_float4(0.f, 0.f, 0.f, 0.f);
}

__global__ __launch_bounds__(256) void degree_kernel(const int* __restrict__ dst,
                                                     float* __restrict__ deg, int E) {
    int i = blockIdx.x * blockDim.x + threadIdx.x;
    if (i < E) atomicAdd(&deg[dst[i]], 1.0f);   // returnless f32 atomic, resolves in L2
}

__global__ __launch_bounds__(256) void dinv_kernel(float* __restrict__ deg_dinv, int N) {
    int i = blockIdx.x * blockDim.x + threadIdx.x;
    if (i < N) {
        float d = fmaxf(deg_dinv[i], 1.0f);
        deg_dinv[i] = 1.0f / sqrtf(d);          // in-place: deg -> dinv
    }
}

// One wave per edge: lanes = feature columns. Gather x[src]*dinv[src] (128B line),
// atomic-add into agg[dst] (128B line). Edge indices forced uniform -> scalar loads.
__global__ __launch_bounds__(256) void scatter_kernel(const float* __restrict__ x,
                                                      const float* __restrict__ dinv,
                                                      float* __restrict__ agg,
                                                      const int* __restrict__ src,
                                                      const int* __restrict__ dst,
                                                      int E) {
    const int lane = threadIdx.x & 31;
    int e = blockIdx.x * (blockDim.x >> 5) + (threadIdx.x >> 5);   // wave-uniform
    if (e >= E) return;                                            // whole wave exits together
    e = __builtin_amdgcn_readfirstlane(e);
    if (lane == 0) {
        // stream the edge list ahead through L2 (gfx1250 global_prefetch_b8;
        // speculative: OOB addresses are silently dropped)
        __builtin_prefetch(src + e + 4096, 0, 3);
        __builtin_prefetch(dst + e + 4096, 0, 3);
    }
    int s = __builtin_amdgcn_readfirstlane(src[e]);
    int t = __builtin_amdgcn_readfirstlane(dst[e]);
    float dv = dinv[s];                                            // scalar (uniform) load
    float v  = x[s * D_FEAT + lane] * dv;                          // coalesced 128B gather
    atomicAdd(&agg[t * D_FEAT + lane], v);                         // coalesced 128B atomic
}

// y = x + dinv[node] * agg   (vectorized float4; 8 float4 per node since D=32)
__global__ __launch_bounds__(256) void axpy_kernel(const float4* __restrict__ x,
                                                   const float4* __restrict__ agg,
                                                   const float* __restrict__ dinv,
                                                   float4* __restrict__ y, int n4) {
    int i = blockIdx.x * blockDim.x + threadIdx.x;
    if (i >= n4) return;
    float dv = dinv[i >> 3];
    float4 a = agg[i], b = x[i];
    y[i] = make_float4(fmaf(dv, a.x, b.x), fmaf(dv, a.y, b.y),
                       fmaf(dv, a.z, b.z), fmaf(dv, a.w, b.w));
}

// out = c0*F + c1*(F - dinv*agg)
__global__ __launch_bounds__(256) void final_kernel(const float4* __restrict__ F,
                                                    const float4* __restrict__ agg,
                                                    const float* __restrict__ dinv,
                                                    const float* __restrict__ w,
                                                    float4* __restrict__ out, int n4) {
    int i = blockIdx.x * blockDim.x + threadIdx.x;
    if (i >= n4) return;
    float w0 = fmaxf(w[0], 0.f), w1 = fmaxf(w[1], 0.f), w2 = fmaxf(w[2], 0.f);
    float c0 = 0.25f * w0;
    float c1 = 0.5f * w1 + 0.25f * w2;
    float dv = dinv[i >> 3];
    float4 f = F[i], a = agg[i];
    float4 lx = make_float4(f.x - dv * a.x, f.y - dv * a.y,
                            f.z - dv * a.z, f.w - dv * a.w);
    out[i] = make_float4(fmaf(c0, f.x, c1 * lx.x), fmaf(c0, f.y, c1 * lx.y),
                         fmaf(c0, f.z, c1 * lx.z), fmaf(c0, f.w, c1 * lx.w));
}

extern "C" void kernel_launch(void* const* d_in, const int* in_sizes, int n_in,
                              void* d_out, int out_size, void* d_ws, size_t ws_size,
                              hipStream_t stream) {
    const float* feat = (const float*)d_in[0];   // (N, 32) f32
    const float* w    = (const float*)d_in[1];   // (3,)   f32
    const int*   src  = (const int*)d_in[2];     // (E,)   int
    const int*   dst  = (const int*)d_in[3];     // (E,)   int

    const int N  = in_sizes[0] / D_FEAT;
    const int E  = in_sizes[2];
    const int ND = N * D_FEAT;

    // ws layout (floats): [deg/dinv : N][agg : ND][F : ND]  (~13 MB, all 16B-aligned)
    float* deg = (float*)d_ws;
    float* agg = deg + N;
    float* F   = agg + ND;
    float* out = (float*)d_out;

    const int B = 256;
    auto blocks = [](long long n, int b) { return (int)((n + b - 1) / b); };

    const int n4      = ND / 4;                  // ND divisible by 4 (D=32)
    const int z4      = (N + ND) / 4;            // deg+agg contiguous zero
    const int sgrid   = blocks((long long)E * 32, B);

    // degree -> dinv
    zero4_kernel<<<blocks(z4, B), B, 0, stream>>>((float4*)deg, z4);
    degree_kernel<<<blocks(E, B), B, 0, stream>>>(dst, deg, E);
    dinv_kernel<<<blocks(N, B), B, 0, stream>>>(deg, N);
    const float* dinv = deg;

    // f1 = feat + Ahat(feat)
    scatter_kernel<<<sgrid, B, 0, stream>>>(feat, dinv, agg, src, dst, E);
    axpy_kernel<<<blocks(n4, B), B, 0, stream>>>((const float4*)feat, (const float4*)agg,
                                                 dinv, (float4*)F, n4);

    // f2 = f1 + Ahat(f1)   (in-place on F)
    zero4_kernel<<<blocks(n4, B), B, 0, stream>>>((float4*)agg, n4);
    scatter_kernel<<<sgrid, B, 0, stream>>>(F, dinv, agg, src, dst, E);
    axpy_kernel<<<blocks(n4, B), B, 0, stream>>>((const float4*)F, (const float4*)agg,
                                                 dinv, (float4*)F, n4);

    // out = c0*f2 + c1*(f2 - Ahat(f2))
    zero4_kernel<<<blocks(n4, B), B, 0, stream>>>((float4*)agg, n4);
    scatter_kernel<<<sgrid, B, 0, stream>>>(F, dinv, agg, src, dst, E);
    final_kernel<<<blocks(n4, B), B, 0, stream>>>((const float4*)F, (const float4*)agg,
                                                  dinv, w, (float4*)out, n4);
}